// MessagePassing_72610717106528
// MI455X (gfx1250) — compile-verified
//
#include <hip/hip_runtime.h>
#include <stdint.h>

// Scatter-mean message passing:
//   out[i,:] = (1/max(deg_in(i),1)) * sum_{e: tgt[e]==i} x[src[e],:]
// N=100000, F=64, E=1600000 (F hardcoded to 64 per reference).
//
// MI455X reasoning: x (25.6MB) and out (25.6MB) are L2-resident (192MB L2),
// so HBM traffic (~64MB ~ 3us @ 23.3TB/s) is irrelevant. Bottleneck is L2
// fp32 atomic throughput: E*F = 102.4M global_atomic_add_f32. Design goals:
//  - native HW fp atomics (unsafeAtomicAdd -> global_atomic_add_f32)
//  - COLUMN feature layout: each atomic instruction across a 16-lane group
//    hits 16 consecutive dwords = exactly one 64B line per instruction
//    (4 line-transactions per 256B edge row instead of 16)
//  - index staging via CDNA5 async global->LDS (ASYNCcnt) with wave-private
//    LDS slices (no workgroup barrier needed)
// No matmul structure exists in scatter-mean, so WMMA is inapplicable; the
// gfx1250-specific path exercised is the async-tensor/ASYNCcnt machinery.

#define FEAT 64
#define BLOCKT 256          // 8 waves of 32
#define EDGES_PER_BLOCK 256 // 32 edges per wave

__device__ __forceinline__ unsigned lds_off(const void* p) {
  // Generic pointer to LDS: low 32 bits are the LDS byte offset on gfx1250.
  return (unsigned)(uintptr_t)p;
}

__device__ __forceinline__ void async_gather_b32(unsigned lds_addr, const int* gaddr) {
  // GLOBAL_LOAD_ASYNC_TO_LDS_B32, GV mode: each lane copies 4 bytes from the global address in gaddr to the LDS address in lds_addr.
  asm volatile("global_load_async_to_lds_b32 %0, %1, off"
               :: "v"(lds_addr), "v"(gaddr)
               : "memory");
}

__device__ __forceinline__ void wait_async0() {
  asm volatile("s_wait_asynccnt 0" ::: "memory");
}

__device__ __forceinline__ void scatter_edge(const float* __restrict__ x,
                                             float* __restrict__ out,
                                             float* __restrict__ counts,
                                             int s, int t, int chunk) {
  const float* xrow = x + (size_t)s * FEAT;
  float*       orow = out + (size_t)t * FEAT;
  // Column layout: lane `chunk` owns features {chunk, chunk+16, chunk+32, chunk+48}.
  // Each of the 4 atomic (and load) instructions across the 16-lane group
  // touches one contiguous 64B cacheline.
#pragma unroll
  for (int k = 0; k < 4; ++k) {
    const int f = (k << 4) + chunk;
    unsafeAtomicAdd(orow + f, xrow[f]);
  }
  if (chunk == 0) {
    unsafeAtomicAdd(counts + t, 1.0f);
  }
}

__global__ __launch_bounds__(BLOCKT) void mp_edge_scatter(
    const float* __restrict__ x,
    const int* __restrict__ src,
    const int* __restrict__ tgt,
    float* __restrict__ out,
    float* __restrict__ counts,
    int E)
{
  __shared__ int lds_src[EDGES_PER_BLOCK];
  __shared__ int lds_tgt[EDGES_PER_BLOCK];

  const int tid   = threadIdx.x;
  const int lane  = tid & 31;
  const int wave  = tid >> 5;              // 0..7
  const int tile  = blockIdx.x * EDGES_PER_BLOCK;
  const int slot  = (wave << 5) + lane;    // this lane's staging slot
  const int eidx  = tile + slot;

  // --- Stage this wave's 32 edge indices into its private LDS slice (async).
  if (eidx < E) {
    async_gather_b32(lds_off(&lds_src[slot]), src + eidx);
    async_gather_b32(lds_off(&lds_tgt[slot]), tgt + eidx);
  }
  wait_async0();  // wave-private slice: ASYNCcnt fence only, no barrier

  // --- Process: 16 lanes per edge (2 edges per wave per pass).
  const int sub   = lane >> 4;   // which of 2 edges this half-wave handles
  const int chunk = lane & 15;   // feature column within the row

  if (tile + EDGES_PER_BLOCK <= E) {
    // Fast path: full tile, no bound checks in the hot loop.
#pragma unroll 4
    for (int p = 0; p < 16; ++p) {
      const int ls = (wave << 5) + (p << 1) + sub;
      scatter_edge(x, out, counts, lds_src[ls], lds_tgt[ls], chunk);
    }
  } else {
    for (int p = 0; p < 16; ++p) {
      const int ls = (wave << 5) + (p << 1) + sub;
      if (tile + ls < E) {
        scatter_edge(x, out, counts, lds_src[ls], lds_tgt[ls], chunk);
      }
    }
  }
}

__global__ __launch_bounds__(BLOCKT) void mp_normalize(
    float* __restrict__ out,
    const float* __restrict__ counts,
    int N)
{
  const int i = blockIdx.x * blockDim.x + threadIdx.x;   // one float4 chunk
  const int total = N * (FEAT / 4);
  if (i >= total) return;
  const int node = i / (FEAT / 4);
  const float c = counts[node];
  const float r = 1.0f / fmaxf(c, 1.0f);
  float4* p = (float4*)out + i;
  float4 v = *p;
  v.x *= r; v.y *= r; v.z *= r; v.w *= r;
  *p = v;
}

extern "C" void kernel_launch(void* const* d_in, const int* in_sizes, int n_in,
                              void* d_out, int out_size, void* d_ws, size_t ws_size,
                              hipStream_t stream) {
  const float* x    = (const float*)d_in[0];
  const int*   eidx = (const int*)d_in[1];

  const int NF = in_sizes[0];      // N * 64
  const int N  = NF / FEAT;
  const int E  = in_sizes[1] / 2;

  const int* src = eidx;           // row 0: source j
  const int* tgt = eidx + E;       // row 1: target i

  float* out    = (float*)d_out;
  float* counts = (float*)d_ws;    // N floats of scratch

  // Zero accumulators every call (atomic accumulation requires clean start).
  hipMemsetAsync(out, 0, sizeof(float) * (size_t)NF, stream);
  hipMemsetAsync(counts, 0, sizeof(float) * (size_t)N, stream);

  const int blocksE = (E + EDGES_PER_BLOCK - 1) / EDGES_PER_BLOCK;
  mp_edge_scatter<<<blocksE, BLOCKT, 0, stream>>>(x, src, tgt, out, counts, E);

  const int chunks  = N * (FEAT / 4);
  const int blocksN = (chunks + BLOCKT - 1) / BLOCKT;
  mp_normalize<<<blocksN, BLOCKT, 0, stream>>>(out, counts, N);
}